// VIMECorruption_7438883357301
// MI455X (gfx1250) — compile-verified
//
#include <hip/hip_runtime.h>
#include <hip/hip_bf16.h>
#include <stdint.h>

// VIME corruption: out = mask ? (swap ? x[row, perm] : x) : 0.1*noise
// B=16384 rows, F=2048 features. One block per row; row of x staged in LDS
// via gfx1250 async-to-LDS (ASYNCcnt) path; streaming operands use TH=NT.

#define VIME_F 2048
#define VIME_NOISE_SCALE 0.10f

typedef float  f32x4 __attribute__((ext_vector_type(4)));
typedef int    i32x4 __attribute__((ext_vector_type(4)));

__global__ __launch_bounds__(256) void vime_corruption_kernel(
    const float*         __restrict__ x,
    const unsigned char* __restrict__ mask,       // keep-mask, 1 byte per elem
    const unsigned char* __restrict__ swap_mask,  // 1 byte per elem
    const int*           __restrict__ perms,
    const float*         __restrict__ noise,
    float*               __restrict__ out)
{
    __shared__ float srow[VIME_F];  // 8 KB: one full row of x

    const int     tid  = threadIdx.x;            // 0..255
    const int64_t base = (int64_t)blockIdx.x * VIME_F;
    const float*  xrow = x + base;

    // ---- Stage the row into LDS --------------------------------------------
#if __has_builtin(__builtin_amdgcn_global_load_async_to_lds_b128)
    {
        typedef __attribute__((address_space(1))) i32x4 gvec_t;  // __device__
        typedef __attribute__((address_space(3))) i32x4 lvec_t;  // __shared__
        gvec_t* g0 = (gvec_t*)(xrow + 4 * tid);
        gvec_t* g1 = (gvec_t*)(xrow + 1024 + 4 * tid);
        lvec_t* l0 = (lvec_t*)(&srow[4 * tid]);
        lvec_t* l1 = (lvec_t*)(&srow[1024 + 4 * tid]);
        __builtin_amdgcn_global_load_async_to_lds_b128(g0, l0, 0, 0);
        __builtin_amdgcn_global_load_async_to_lds_b128(g1, l1, 0, 0);
#if __has_builtin(__builtin_amdgcn_s_wait_asynccnt)
        __builtin_amdgcn_s_wait_asynccnt(0);
#else
        asm volatile("s_wait_asynccnt 0x0" ::: "memory");
#endif
    }
#else
    {
        // Fallback: b128 global load -> ds_store_b128
        f32x4 v0 = __builtin_nontemporal_load((const f32x4*)(xrow) + tid);
        f32x4 v1 = __builtin_nontemporal_load((const f32x4*)(xrow) + 256 + tid);
        ((f32x4*)srow)[tid]       = v0;
        ((f32x4*)srow)[256 + tid] = v1;
    }
#endif
    __syncthreads();  // all 8 waves' row data visible in LDS

    // ---- Process two coalesced float4 chunks per thread --------------------
#pragma unroll
    for (int c = 0; c < 2; ++c) {
        const int     j   = c * 1024 + 4 * tid;      // element offset in row
        const int64_t off = base + j;

        // Streaming, single-use data: non-temporal hint (TH=NT).
        i32x4    p4 = __builtin_nontemporal_load((const i32x4*)(perms + off));
        f32x4    n4 = __builtin_nontemporal_load((const f32x4*)(noise + off));
        uint32_t m4 = __builtin_nontemporal_load((const uint32_t*)(mask + off));
        uint32_t s4 = __builtin_nontemporal_load((const uint32_t*)(swap_mask + off));

        // Identity values come from LDS (ds_load_b128), not a 2nd global read.
        f32x4 xv = *(const f32x4*)(&srow[j]);

        f32x4 r;
#pragma unroll
        for (int k = 0; k < 4; ++k) {
            const bool  sw  = ((s4 >> (8 * k)) & 0xffu) != 0u;
            const bool  mk  = ((m4 >> (8 * k)) & 0xffu) != 0u;
            const int   p   = p4[k] & (VIME_F - 1);   // free OOB safety
            const float gat = srow[p];                // ds_load_b32 gather
            const float val = sw ? gat : xv[k];
            r[k] = mk ? val : n4[k] * VIME_NOISE_SCALE;
        }
        __builtin_nontemporal_store(r, (f32x4*)(out + off));
    }
}

extern "C" void kernel_launch(void* const* d_in, const int* in_sizes, int n_in,
                              void* d_out, int out_size, void* d_ws, size_t ws_size,
                              hipStream_t stream) {
    // setup_inputs order: x(f32), mask(bool/u8), swap_mask(bool/u8),
    //                     perms(i32), noise(f32)
    const float*         x         = (const float*)d_in[0];
    const unsigned char* mask      = (const unsigned char*)d_in[1];
    const unsigned char* swap_mask = (const unsigned char*)d_in[2];
    const int*           perms     = (const int*)d_in[3];
    const float*         noise     = (const float*)d_in[4];
    float*               out       = (float*)d_out;

    const int B = in_sizes[0] / VIME_F;  // 16384

    vime_corruption_kernel<<<B, 256, 0, stream>>>(x, mask, swap_mask, perms,
                                                  noise, out);
}